// DecoderSplattingCUDA_52974126629296
// MI455X (gfx1250) — compile-verified
//
#include <hip/hip_runtime.h>
#include <math.h>

typedef __attribute__((ext_vector_type(2))) float v2f;
typedef __attribute__((ext_vector_type(8))) float v8f;
typedef __attribute__((ext_vector_type(4))) float f4;
typedef __attribute__((ext_vector_type(4))) int   i4;

#define SH_C0 0.28209479177387814f
#define SH_C1 0.4886025119029199f
#define ZNEAR 0.1f
#define ZFAR  1000.0f
#define CHUNK 256
#define SORT_N 2048

#if defined(__has_builtin)
#if __has_builtin(__builtin_amdgcn_global_load_async_to_lds_b128)
#define HAVE_ASYNC_LDS 1
#endif
#endif

// ------------------------- 4x4 inverse (adjugate) -------------------------
__device__ inline void inv4x4(const float* m, float* inv) {
  inv[0]  =  m[5]*m[10]*m[15] - m[5]*m[11]*m[14] - m[9]*m[6]*m[15] + m[9]*m[7]*m[14] + m[13]*m[6]*m[11] - m[13]*m[7]*m[10];
  inv[4]  = -m[4]*m[10]*m[15] + m[4]*m[11]*m[14] + m[8]*m[6]*m[15] - m[8]*m[7]*m[14] - m[12]*m[6]*m[11] + m[12]*m[7]*m[10];
  inv[8]  =  m[4]*m[9]*m[15]  - m[4]*m[11]*m[13] - m[8]*m[5]*m[15] + m[8]*m[7]*m[13] + m[12]*m[5]*m[11] - m[12]*m[7]*m[9];
  inv[12] = -m[4]*m[9]*m[14]  + m[4]*m[10]*m[13] + m[8]*m[5]*m[14] - m[8]*m[6]*m[13] - m[12]*m[5]*m[10] + m[12]*m[6]*m[9];
  inv[1]  = -m[1]*m[10]*m[15] + m[1]*m[11]*m[14] + m[9]*m[2]*m[15] - m[9]*m[3]*m[14] - m[13]*m[2]*m[11] + m[13]*m[3]*m[10];
  inv[5]  =  m[0]*m[10]*m[15] - m[0]*m[11]*m[14] - m[8]*m[2]*m[15] + m[8]*m[3]*m[14] + m[12]*m[2]*m[11] - m[12]*m[3]*m[10];
  inv[9]  = -m[0]*m[9]*m[15]  + m[0]*m[11]*m[13] + m[8]*m[1]*m[15] - m[8]*m[3]*m[13] - m[12]*m[1]*m[11] + m[12]*m[3]*m[9];
  inv[13] =  m[0]*m[9]*m[14]  - m[0]*m[10]*m[13] - m[8]*m[1]*m[14] + m[8]*m[2]*m[13] + m[12]*m[1]*m[10] - m[12]*m[2]*m[9];
  inv[2]  =  m[1]*m[6]*m[15]  - m[1]*m[7]*m[14]  - m[5]*m[2]*m[15] + m[5]*m[3]*m[14] + m[13]*m[2]*m[7]  - m[13]*m[3]*m[6];
  inv[6]  = -m[0]*m[6]*m[15]  + m[0]*m[7]*m[14]  + m[4]*m[2]*m[15] - m[4]*m[3]*m[14] - m[12]*m[2]*m[7]  + m[12]*m[3]*m[6];
  inv[10] =  m[0]*m[5]*m[15]  - m[0]*m[7]*m[13]  - m[4]*m[1]*m[15] + m[4]*m[3]*m[13] + m[12]*m[1]*m[7]  - m[12]*m[3]*m[5];
  inv[14] = -m[0]*m[5]*m[14]  + m[0]*m[6]*m[13]  + m[4]*m[1]*m[14] - m[4]*m[2]*m[13] - m[12]*m[1]*m[6]  + m[12]*m[2]*m[5];
  inv[3]  = -m[1]*m[6]*m[11]  + m[1]*m[7]*m[10]  + m[5]*m[2]*m[11] - m[5]*m[3]*m[10] - m[9]*m[2]*m[7]   + m[9]*m[3]*m[6];
  inv[7]  =  m[0]*m[6]*m[11]  - m[0]*m[7]*m[10]  - m[4]*m[2]*m[11] + m[4]*m[3]*m[10] + m[8]*m[2]*m[7]   - m[8]*m[3]*m[6];
  inv[11] = -m[0]*m[5]*m[11]  + m[0]*m[7]*m[9]   + m[4]*m[1]*m[11] - m[4]*m[3]*m[9]  - m[8]*m[1]*m[7]   + m[8]*m[3]*m[5];
  inv[15] =  m[0]*m[5]*m[10]  - m[0]*m[6]*m[9]   - m[4]*m[1]*m[10] + m[4]*m[2]*m[9]  + m[8]*m[1]*m[6]   - m[8]*m[2]*m[5];
  float det = m[0]*inv[0] + m[1]*inv[4] + m[2]*inv[8] + m[3]*inv[12];
  det = 1.0f / det;
  for (int i = 0; i < 16; ++i) inv[i] *= det;
}

// ------------------- Kernel 1: per-camera matrices -------------------
// cambuf per camera (stride 32 floats):
// [0..8]=w2c R, [9..11]=w2c t, [12..15]=fx,fy,cx,cy, [16..18]=cam_pos
__global__ void k_prep_cams(const float* __restrict__ ctx, const float* __restrict__ tpos,
                            const float* __restrict__ intr, float* __restrict__ cambuf,
                            int bv, int v) {
  int c = blockIdx.x * blockDim.x + threadIdx.x;
  if (c >= bv) return;
  int bi = c / v;
  float invc[16], extr[16], w2c[16];
  inv4x4(ctx + 16 * bi, invc);
  const float* tp = tpos + 16 * c;
  for (int i = 0; i < 4; ++i)
    for (int j = 0; j < 4; ++j) {
      float s = 0.f;
      for (int k = 0; k < 4; ++k) s += invc[i*4+k] * tp[k*4+j];
      extr[i*4+j] = s;
    }
  inv4x4(extr, w2c);
  float* cb = cambuf + 32 * c;
  cb[0]=w2c[0]; cb[1]=w2c[1]; cb[2]=w2c[2];
  cb[3]=w2c[4]; cb[4]=w2c[5]; cb[5]=w2c[6];
  cb[6]=w2c[8]; cb[7]=w2c[9]; cb[8]=w2c[10];
  cb[9]=w2c[3]; cb[10]=w2c[7]; cb[11]=w2c[11];
  const float* K = intr + 9 * c;
  cb[12]=K[0]; cb[13]=K[4]; cb[14]=K[2]; cb[15]=K[5]; // normalization by W,H cancels
  cb[16]=extr[3]; cb[17]=extr[7]; cb[18]=extr[11];
}

// ------------------- Kernel 2: per-gaussian preprocess -------------------
__global__ void k_prep_gauss(const float* __restrict__ means1, const float* __restrict__ means2,
                             const float* __restrict__ cov1,  const float* __restrict__ cov2,
                             const float* __restrict__ sh1,   const float* __restrict__ sh2,
                             const float* __restrict__ op1,   const float* __restrict__ op2,
                             const float* __restrict__ cambuf,
                             float* __restrict__ recs, float* __restrict__ zkey,
                             int bv, int v, int G, int Gh) {
  int id = blockIdx.x * blockDim.x + threadIdx.x;
  if (id >= bv * G) return;
  int cam = id / G, g = id % G;
  int bi = cam / v;
  const float *mean, *cov, *sh; float op;
  if (g < Gh) { int o = bi*Gh + g;        mean = means1+3*o; cov = cov1+9*o; sh = sh1+12*o; op = op1[o]; }
  else        { int o = bi*Gh + (g - Gh); mean = means2+3*o; cov = cov2+9*o; sh = sh2+12*o; op = op2[o]; }

  const float* cb = cambuf + 32 * cam;
  float R[9], t3[3];
  for (int i = 0; i < 9; ++i) R[i] = cb[i];
  t3[0]=cb[9]; t3[1]=cb[10]; t3[2]=cb[11];
  float fx=cb[12], fy=cb[13], cx=cb[14], cy=cb[15];

  float m0=mean[0], m1=mean[1], m2=mean[2];
  float X = R[0]*m0 + R[1]*m1 + R[2]*m2 + t3[0];
  float Y = R[3]*m0 + R[4]*m1 + R[5]*m2 + t3[1];
  float Z = R[6]*m0 + R[7]*m1 + R[8]*m2 + t3[2];
  float zc = fmaxf(Z, 1e-6f);
  float u  = fx * X / zc + cx;
  float vv = fy * Y / zc + cy;

  float T9[9], CC[9];
  for (int i = 0; i < 3; ++i)
    for (int k = 0; k < 3; ++k) {
      float s = 0.f;
      for (int j = 0; j < 3; ++j) s += R[i*3+j] * cov[j*3+k];
      T9[i*3+k] = s;
    }
  for (int i = 0; i < 3; ++i)
    for (int l = 0; l < 3; ++l) {
      float s = 0.f;
      for (int k = 0; k < 3; ++k) s += T9[i*3+k] * R[l*3+k];
      CC[i*3+l] = s;
    }
  float j00 = fx/zc, j02 = -fx*X/(zc*zc), j11 = fy/zc, j12 = -fy*Y/(zc*zc);
  float a00 = j00*CC[0] + j02*CC[6], a01 = j00*CC[1] + j02*CC[7], a02 = j00*CC[2] + j02*CC[8];
  float a11 = j11*CC[4] + j12*CC[7], a12 = j11*CC[5] + j12*CC[8];
  float c00 = a00*j00 + a02*j02 + 0.3f;
  float c01 = a01*j11 + a02*j12;
  float c11 = a11*j11 + a12*j12 + 0.3f;
  float det = fmaxf(c00*c11 - c01*c01, 1e-12f);
  float ia = c11/det, ib = -c01/det, ic = c00/det;

  float dx = m0 - cb[16], dy = m1 - cb[17], dz = m2 - cb[18];
  float nn = sqrtf(dx*dx + dy*dy + dz*dz);
  dx /= nn; dy /= nn; dz /= nn;
  float col[3];
  for (int ch = 0; ch < 3; ++ch) {
    float c = SH_C0 * sh[ch*4+0]
            - SH_C1 * dy * sh[ch*4+1]
            + SH_C1 * dz * sh[ch*4+2]
            - SH_C1 * dx * sh[ch*4+3];
    col[ch] = fmaxf(c + 0.5f, 0.0f);
  }
  bool valid = (Z > ZNEAR) && (Z < ZFAR);

  float* r = recs + (size_t)(cam*G + g) * 12;
  r[0] = -0.5f * ia;
  r[1] = -0.5f * ic;
  r[2] = -ib;
  r[3] = ia*u + ib*vv;
  r[4] = ic*vv + ib*u;
  r[5] = -0.5f*(ia*u*u + ic*vv*vv) - ib*u*vv;
  r[6] = valid ? op : 0.0f;
  r[7] = col[0]; r[8] = col[1]; r[9] = col[2];
  r[10] = Z; r[11] = 0.f;
  zkey[cam*G + g] = Z;
}

// ------------------- Kernel 3: bitonic depth sort (per camera) -------------------
__global__ void k_sort(const float* __restrict__ zkey, int* __restrict__ order, int G) {
  __shared__ float key[SORT_N];
  __shared__ int   idx[SORT_N];
  int cam = blockIdx.x, tid = threadIdx.x, nt = blockDim.x;
  for (int i = tid; i < SORT_N; i += nt) {
    key[i] = (i < G) ? zkey[cam*G + i] : __builtin_huge_valf();
    idx[i] = i;
  }
  __syncthreads();
  for (int k = 2; k <= SORT_N; k <<= 1) {
    for (int j = k >> 1; j > 0; j >>= 1) {
      for (int i = tid; i < SORT_N; i += nt) {
        int ixj = i ^ j;
        if (ixj > i) {
          bool up = ((i & k) == 0);
          float ki = key[i], kj = key[ixj];
          bool sw = up ? (ki > kj) : (ki < kj);
          if (sw) {
            key[i] = kj; key[ixj] = ki;
            int ti = idx[i]; idx[i] = idx[ixj]; idx[ixj] = ti;
          }
        }
      }
      __syncthreads();
    }
  }
  for (int i = tid; i < G; i += nt) order[cam*G + i] = idx[i];
}

// ------------------- Kernel 4: gather sorted records -------------------
__global__ void k_gather(const float* __restrict__ recs, const int* __restrict__ order,
                         float* __restrict__ srecs, int G) {
  int id = blockIdx.x * blockDim.x + threadIdx.x;
  int cam = id / G, i = id % G;
  int src = order[cam*G + i];
  const f4* a = (const f4*)(recs + (size_t)(cam*G + src) * 12);
  f4* b = (f4*)(srecs + (size_t)(cam*G + i) * 12);
  b[0] = a[0]; b[1] = a[1]; b[2] = a[2];
}

// -------- async global->LDS staging of one 48B record (3 x b128) --------
// builtin signature (from compiler diagnostics): (int4 AS1*, int4 AS3*, int, int)
__device__ inline void stage_rec_async(const float* src, float* dst) {
#ifdef HAVE_ASYNC_LDS
  __builtin_amdgcn_global_load_async_to_lds_b128(
      (__attribute__((address_space(1))) i4*)(src),
      (__attribute__((address_space(3))) i4*)(dst), 0, 0);
  __builtin_amdgcn_global_load_async_to_lds_b128(
      (__attribute__((address_space(1))) i4*)(src + 4),
      (__attribute__((address_space(3))) i4*)(dst + 4), 0, 0);
  __builtin_amdgcn_global_load_async_to_lds_b128(
      (__attribute__((address_space(1))) i4*)(src + 8),
      (__attribute__((address_space(3))) i4*)(dst + 8), 0, 0);
#else
  const f4* s = (const f4*)src;
  f4* d = (f4*)dst;
  d[0] = s[0]; d[1] = s[1]; d[2] = s[2];
#endif
}

__device__ inline void wait_async_lds() {
#ifdef HAVE_ASYNC_LDS
#if __has_builtin(__builtin_amdgcn_s_wait_asynccnt)
  __builtin_amdgcn_s_wait_asynccnt(0);
#else
  asm volatile("s_wait_asynccnt 0x0" ::: "memory");
#endif
#endif
}

// ------------------- Kernel 5: tiled raster w/ WMMA power-GEMM -------------------
// Per wave: 16-pixel x-strip.  Double-buffered async LDS staging of 256-gaussian
// chunks; per 16-gaussian batch: P[16px,16g] = F[16px,8] @ H[8,16g] via
// 2x v_wmma_f32_16x16x4_f32, then segmented transmittance scan + color FMA.
__global__ __launch_bounds__(256) void k_raster(const float* __restrict__ srecs,
                                                const float* __restrict__ bg,
                                                float* __restrict__ out,
                                                int G, int H, int W) {
  __shared__ float lds[2 * CHUNK * 12];
  int cam  = blockIdx.z;
  int x0   = blockIdx.x * 16;
  int yy   = blockIdx.y * 8 + (threadIdx.x >> 5);
  int lane = threadIdx.x & 31;
  int col16 = lane & 15;
  int half  = lane >> 4;

  float px = (float)x0 + (float)col16 + 0.5f;   // A-frag row M = lane%16
  float py = (float)yy + 0.5f;

  // A-frag (f32 16x4): lane l: M=l%16, VGPR p -> K = p + 2*(l/16)
  v2f A0, A1;
  if (half == 0) { A0.x = px*px; A0.y = py*py; A1.x = py;  A1.y = 1.0f; }
  else           { A0.x = px*py; A0.y = px;    A1.x = 0.f; A1.y = 0.f;  }

  float accR[8], accG[8], accB[8], Tc[8];
#pragma unroll
  for (int r = 0; r < 8; ++r) { accR[r]=accG[r]=accB[r]=0.f; Tc[r]=1.f; }

  const float* base = srecs + (size_t)cam * G * 12;
  const int NC = G / CHUNK;

  // process one staged chunk (16-gaussian batches -> 2 WMMAs each)
  auto process = [&](const float* buf) {
    for (int b16 = 0; b16 < CHUNK; b16 += 16) {
      const float* gr = buf + (b16 + col16) * 12;
      float h0=gr[0], h1=gr[1], h2=gr[2], h3=gr[3], h4=gr[4], h5=gr[5];
      float gop=gr[6], gcr=gr[7], gcg=gr[8], gcb=gr[9];

      // B-frag (f32 4x16): lane l: N=l%16, VGPR p -> K = p + 2*(l/16)
      v2f B0, B1;
      if (half == 0) { B0.x = h0; B0.y = h1; B1.x = h4;  B1.y = h5;  }
      else           { B0.x = h2; B0.y = h3; B1.x = 0.f; B1.y = 0.f; }

      v8f P = {0.f,0.f,0.f,0.f,0.f,0.f,0.f,0.f};
#if __has_builtin(__builtin_amdgcn_wmma_f32_16x16x4_f32)
      P = __builtin_amdgcn_wmma_f32_16x16x4_f32(false, A1, false, B1, (short)0, P, false, false);
      P = __builtin_amdgcn_wmma_f32_16x16x4_f32(false, A0, false, B0, (short)0, P, false, false);
#else
#pragma unroll
      for (int r = 0; r < 8; ++r) {
        float Xm = (float)x0 + (float)(r + 8*half) + 0.5f;
        P[r] = h0*Xm*Xm + h1*py*py + h2*Xm*py + h3*Xm + h4*py + h5;
      }
#endif

#pragma unroll
      for (int r = 0; r < 8; ++r) {
        float power = P[r];
        float alpha = fminf(0.99f, gop * __expf(power));
        if (power > 0.0f || alpha < (1.0f/255.0f)) alpha = 0.0f;
        float s = 1.0f - alpha;            // segmented inclusive product scan
#pragma unroll
        for (int d = 1; d < 16; d <<= 1) {
          float tt = __shfl_up(s, d, 16);
          if (col16 >= d) s *= tt;
        }
        float e = __shfl_up(s, 1, 16);
        if (col16 == 0) e = 1.0f;
        float w = alpha * Tc[r] * e;       // alpha * T_prev
        accR[r] = fmaf(w, gcr, accR[r]);
        accG[r] = fmaf(w, gcg, accG[r]);
        accB[r] = fmaf(w, gcb, accB[r]);
        Tc[r] *= __shfl(s, 15, 16);
      }
    }
  };

  // prologue: stage chunk 0 into buffer 0
  {
    int t = threadIdx.x;
    stage_rec_async(base + (size_t)t * 12, lds + t * 12);
    wait_async_lds();
  }
  __syncthreads();

  for (int ci = 0; ci < NC; ++ci) {
    float* cur = lds + (ci & 1) * (CHUNK * 12);
    if (ci + 1 < NC) {  // overlap next chunk's global->LDS DMA with compute
      int t = threadIdx.x;
      float* nxt = lds + ((ci + 1) & 1) * (CHUNK * 12);
      stage_rec_async(base + (size_t)(ci + 1) * CHUNK * 12 + (size_t)t * 12,
                      nxt + t * 12);
    }
    process(cur);
    wait_async_lds();   // next buffer fully landed in LDS
    __syncthreads();    // everyone done reading cur before it is overwritten
  }

  float bg0 = bg[0], bg1 = bg[1], bg2 = bg[2];
#pragma unroll
  for (int r = 0; r < 8; ++r) {
    float vr = accR[r], vg = accG[r], vb = accB[r];
#pragma unroll
    for (int d = 8; d >= 1; d >>= 1) {
      vr += __shfl_xor(vr, d, 16);
      vg += __shfl_xor(vg, d, 16);
      vb += __shfl_xor(vb, d, 16);
    }
    if (col16 == 0) {
      int m = r + 8 * half;
      float Tf = Tc[r];
      size_t pix = (size_t)yy * W + x0 + m;
      size_t plane = (size_t)H * W;
      out[((size_t)(cam*3 + 0)) * plane + pix] = vr + Tf * bg0;
      out[((size_t)(cam*3 + 1)) * plane + pix] = vg + Tf * bg1;
      out[((size_t)(cam*3 + 2)) * plane + pix] = vb + Tf * bg2;
    }
  }
}

// ------------------------------- launcher -------------------------------
extern "C" void kernel_launch(void* const* d_in, const int* in_sizes, int n_in,
                              void* d_out, int out_size, void* d_ws, size_t ws_size,
                              hipStream_t stream) {
  (void)n_in; (void)ws_size;
  const float* ctx    = (const float*)d_in[0];
  const float* tpos   = (const float*)d_in[1];
  const float* intr   = (const float*)d_in[2];
  const float* means1 = (const float*)d_in[3];
  const float* means2 = (const float*)d_in[4];
  const float* cov1   = (const float*)d_in[5];
  const float* cov2   = (const float*)d_in[6];
  const float* sh1    = (const float*)d_in[7];
  const float* sh2    = (const float*)d_in[8];
  const float* op1    = (const float*)d_in[9];
  const float* op2    = (const float*)d_in[10];
  const float* bg     = (const float*)d_in[11];
  float* out = (float*)d_out;

  int b  = in_sizes[0] / 16;
  int bv = in_sizes[1] / 16;
  int v  = bv / b;
  int Gh = in_sizes[3] / (3 * b);
  int G  = 2 * Gh;
  int HW = out_size / (3 * bv);
  int H  = (int)(sqrt((double)HW) + 0.5);
  int W  = H;  // square images in this workload

  float* wsf    = (float*)d_ws;
  float* cambuf = wsf;                                   // bv*32
  float* zkey   = cambuf + (size_t)bv * 32;              // bv*G
  float* recs   = zkey + (size_t)bv * G;                 // bv*G*12
  float* srecs  = recs + (size_t)bv * G * 12;            // bv*G*12
  int*   order  = (int*)(srecs + (size_t)bv * G * 12);   // bv*G

  k_prep_cams<<<1, 32, 0, stream>>>(ctx, tpos, intr, cambuf, bv, v);
  int n1 = bv * G;
  k_prep_gauss<<<(n1 + 255) / 256, 256, 0, stream>>>(means1, means2, cov1, cov2,
                                                     sh1, sh2, op1, op2,
                                                     cambuf, recs, zkey, bv, v, G, Gh);
  k_sort<<<bv, 1024, 0, stream>>>(zkey, order, G);
  k_gather<<<(n1 + 255) / 256, 256, 0, stream>>>(recs, order, srecs, G);
  dim3 grid(W / 16, H / 8, bv);
  k_raster<<<grid, 256, 0, stream>>>(srecs, bg, out, G, H, W);
}